// SlotClassifier_46497315947152
// MI455X (gfx1250) — compile-verified
//
#include <hip/hip_runtime.h>
#include <hip/hip_bf16.h>
#include <math.h>

typedef __attribute__((ext_vector_type(2))) float v2f;
typedef __attribute__((ext_vector_type(8))) float v8f;

#define B_     16
#define S_     512
#define D_     768
#define L_     20
#define INNER_ 64
#define NEG_   1000000000000.0f

// workspace offsets (in floats)
#define QW_OFF    0u          // 16*512*64  = 524288  floats (2 MB)
#define KW_OFF    524288u     // 524288 floats
#define QK_OFF    1048576u    // 16*512*512 = 4194304 floats (16 MB)
#define DENSE_OFF 5242880u    // 16*40*512  = 327680  floats (1.25 MB)

// ---------------------------------------------------------------------------
// Kernel 1: out1 = x @ w1 + b1  (8192 x 128, K=768) via f32 WMMA 16x16x4,
// then interleaved RoPE split into qw / kw (each (B,S,64)).
// One block = 8 waves = one 16-row M tile, wave w owns N tile [16w,16w+16).
// ---------------------------------------------------------------------------
__global__ __launch_bounds__(256)
void k1_gemm1_rope(const float* __restrict__ x, const float* __restrict__ w1,
                   const float* __restrict__ b1, float* __restrict__ qw,
                   float* __restrict__ kw) {
  const int lane = threadIdx.x & 31;
  const int wid  = threadIdx.x >> 5;     // 0..7 -> N tile
  const int half = lane >> 4;            // 0/1 selects K pair
  const int r16  = lane & 15;
  const int m0   = blockIdx.x * 16;      // flat row over B*S
  const int col  = wid * 16 + r16;       // out1 column 0..127

  const float* arow = x + (size_t)(m0 + r16) * D_;
  __builtin_prefetch(w1 + col, 0, 1);    // global_prefetch_b8: warm w1 in L2

  v8f c = {};
#pragma unroll 4
  for (int k = 0; k < D_; k += 4) {
    v2f a, b;
    a.x = arow[k + 2 * half];            // contiguous pair -> b64 load
    a.y = arow[k + 2 * half + 1];
    b.x = w1[(size_t)(k + 2 * half) * 128 + col];
    b.y = w1[(size_t)(k + 2 * half + 1) * 128 + col];
    c = __builtin_amdgcn_wmma_f32_16x16x4_f32(false, a, false, b,
                                              (short)0, c, false, false);
  }

  const float bb   = b1[col];
  const int   b    = m0 >> 9;            // batch index (tiles never span batches)
  const int   sb   = m0 & 511;           // base sequence position
  const int   d    = col >> 1;           // q/k feature index 0..63
  const int   t    = col >> 2;           // RoPE pair index 0..31
  const float freq = __powf(10000.0f, -2.0f * (float)t / 64.0f);
  float* dst = (col & 1) ? kw : qw;
  const size_t base = (size_t)b * S_ * INNER_ + d;

#pragma unroll
  for (int v = 0; v < 8; ++v) {
    float val  = c[v] + bb;
    float part = __shfl_xor(val, 2, 32); // partner column col^2
    int   s    = sb + v + 8 * half;
    float ang  = (float)s * freq;
    float sn, cs;
    __sincosf(ang, &sn, &cs);
    float res = ((col & 2) == 0) ? (val * cs - part * sn)   // even feature
                                 : (val * cs + part * sn);  // odd feature
    dst[base + (size_t)s * INNER_] = res;
  }
}

// ---------------------------------------------------------------------------
// Kernel 2: qk[b,m,n] = dot(qw[b,m,:], kw[b,n,:]) * 0.125   (K=64)
// One wave per 16x16 tile; 16 WMMA f32 16x16x4 steps.
// ---------------------------------------------------------------------------
__global__ __launch_bounds__(256)
void k2_qk(const float* __restrict__ qw, const float* __restrict__ kw,
           float* __restrict__ qk) {
  const int lane = threadIdx.x & 31;
  const int wid  = threadIdx.x >> 5;
  const int half = lane >> 4;
  const int r16  = lane & 15;
  const int id   = blockIdx.x * 8 + wid;  // 0..16383
  const int b    = id >> 10;
  const int mt   = (id >> 5) & 31;
  const int nt   = id & 31;

  const float* arow = qw + ((size_t)b * S_ + (mt * 16 + r16)) * INNER_;
  const float* brow = kw + ((size_t)b * S_ + (nt * 16 + r16)) * INNER_;

  v8f c = {};
#pragma unroll
  for (int k = 0; k < INNER_; k += 4) {
    v2f a, bb;
    a.x  = arow[k + 2 * half];  a.y  = arow[k + 2 * half + 1];
    bb.x = brow[k + 2 * half];  bb.y = brow[k + 2 * half + 1];
    c = __builtin_amdgcn_wmma_f32_16x16x4_f32(false, a, false, bb,
                                              (short)0, c, false, false);
  }
  const size_t base = (size_t)b * S_ * S_;
#pragma unroll
  for (int v = 0; v < 8; ++v) {
    int m = mt * 16 + v + 8 * half;
    qk[base + (size_t)m * S_ + nt * 16 + r16] = c[v] * 0.125f;
  }
}

// ---------------------------------------------------------------------------
// Kernel 3: dense[b,j,s] = (x[b,s,:] @ w2[:,j] + b2[j]) * 0.5   (j<40)
// 3 waves/block: N padded to 48, B reads clamped, stores masked
// (EXEC stays all-ones through the WMMAs, as the ISA requires).
// ---------------------------------------------------------------------------
__global__ __launch_bounds__(96)
void k3_dense(const float* __restrict__ x, const float* __restrict__ w2,
              const float* __restrict__ b2, float* __restrict__ dense) {
  const int lane = threadIdx.x & 31;
  const int wid  = threadIdx.x >> 5;     // 0..2
  const int half = lane >> 4;
  const int r16  = lane & 15;
  const int m0   = blockIdx.x * 16;
  const int col  = wid * 16 + r16;       // 0..47
  const int colc = col < 39 ? col : 39;  // clamp for loads

  const float* arow = x + (size_t)(m0 + r16) * D_;
  v8f c = {};
#pragma unroll 4
  for (int k = 0; k < D_; k += 4) {
    v2f a, b;
    a.x = arow[k + 2 * half];
    a.y = arow[k + 2 * half + 1];
    b.x = w2[(size_t)(k + 2 * half) * 40 + colc];
    b.y = w2[(size_t)(k + 2 * half + 1) * 40 + colc];
    c = __builtin_amdgcn_wmma_f32_16x16x4_f32(false, a, false, b,
                                              (short)0, c, false, false);
  }
  if (col < 40) {
    const float bb = b2[col];
    const int   b  = m0 >> 9;
    const int   sb = m0 & 511;
#pragma unroll
    for (int v = 0; v < 8; ++v) {
      int s = sb + v + 8 * half;
      dense[((size_t)b * 40 + col) * S_ + s] = (c[v] + bb) * 0.5f;
    }
  }
}

// ---------------------------------------------------------------------------
// Kernel 4: streaming assemble (the HBM-bound phase: 671 MB of stores).
// logits[b,l,m,n] = qk[b,m,n] + dense[b,2l,n] + dense[b,2l+1,m]
// then mask(m), mask(n), causal tril(-1), sigmoid. float4 along n.
// ---------------------------------------------------------------------------
__global__ __launch_bounds__(256)
void k4_assemble(const float* __restrict__ qk, const float* __restrict__ dense,
                 const int* __restrict__ mask, float* __restrict__ logits,
                 float* __restrict__ probs) {
  size_t idx = (size_t)blockIdx.x * 256 + threadIdx.x; // < 20,971,520
  const int n = (int)(idx & 127) * 4;
  size_t rest = idx >> 7;
  const int m = (int)(rest & 511); rest >>= 9;
  const int l = (int)(rest % 20);
  const int b = (int)(rest / 20);

  const float4 q4 = *(const float4*)(qk + ((size_t)b * S_ + m) * S_ + n);
  const float4 dn = *(const float4*)(dense + ((size_t)b * 40 + 2 * l) * S_ + n);
  const float  dm = dense[((size_t)b * 40 + 2 * l + 1) * S_ + m];
  const float  mm = (float)mask[b * S_ + m];
  const int4   mi = *(const int4*)(mask + b * S_ + n);

  const float qv[4]  = {q4.x, q4.y, q4.z, q4.w};
  const float dnv[4] = {dn.x, dn.y, dn.z, dn.w};
  const float mn[4]  = {(float)mi.x, (float)mi.y, (float)mi.z, (float)mi.w};

  float lo[4], pr[4];
#pragma unroll
  for (int i = 0; i < 4; ++i) {
    float v = qv[i] + dnv[i] + dm;
    v = v * mm - NEG_ * (1.0f - mm);
    v = v * mn[i] - NEG_ * (1.0f - mn[i]);
    if (m > n + i) v -= NEG_;            // tril(.., -1)
    lo[i] = v;
    pr[i] = 1.0f / (1.0f + __expf(-v));
  }
  const size_t off = (((size_t)b * L_ + l) * S_ + m) * (size_t)S_ + n;
  *(float4*)(logits + off) = make_float4(lo[0], lo[1], lo[2], lo[3]);
  *(float4*)(probs + off)  = make_float4(pr[0], pr[1], pr[2], pr[3]);
}

// ---------------------------------------------------------------------------
extern "C" void kernel_launch(void* const* d_in, const int* in_sizes, int n_in,
                              void* d_out, int out_size, void* d_ws, size_t ws_size,
                              hipStream_t stream) {
  const float* x    = (const float*)d_in[0];
  const int*   mask = (const int*)  d_in[1];
  const float* w1   = (const float*)d_in[2];
  const float* b1   = (const float*)d_in[3];
  const float* w2   = (const float*)d_in[4];
  const float* b2   = (const float*)d_in[5];

  float* ws    = (float*)d_ws;
  float* qw    = ws + QW_OFF;
  float* kw    = ws + KW_OFF;
  float* qk    = ws + QK_OFF;
  float* dense = ws + DENSE_OFF;

  float* logits = (float*)d_out;
  float* probs  = logits + (size_t)B_ * L_ * S_ * S_;

  // GEMM phases (tiny FLOPs, all WMMA); intermediates stay L2-resident.
  k1_gemm1_rope<<<512, 256, 0, stream>>>(x, w1, b1, qw, kw);   // 8192/16 M tiles
  k3_dense     <<<512,  96, 0, stream>>>(x, w2, b2, dense);
  k2_qk        <<<2048, 256, 0, stream>>>(qw, kw, qk);         // 16384 tiles / 8 waves
  // HBM-bound streaming phase.
  k4_assemble  <<<81920, 256, 0, stream>>>(qk, dense, mask, logits, probs);
}